// IDWT_2D_62534723830104
// MI455X (gfx1250) — compile-verified
//
#include <hip/hip_runtime.h>
#include <hip/hip_bf16.h>
#include <stdint.h>

typedef unsigned int uint32x4 __attribute__((ext_vector_type(4)));
typedef int          int32x4  __attribute__((ext_vector_type(4)));
typedef int          int32x8  __attribute__((ext_vector_type(8)));
typedef float        v4f      __attribute__((ext_vector_type(4)));
typedef float        v2f      __attribute__((ext_vector_type(2)));

#define B_   8
#define C_   48      // groups (output has 4*C_ = 192 channels)
#define H_   128
#define W_   128
#define RPB  8       // rows per block tile

// Input  x:   (8, 192, 128, 128) f32   [192 = 4 subbands x 48 groups]
// Output out: (8, 192, 256, 256) f32
// out[b, 4c+o, 2h+i, 2w+j] = (sum_n x[b, n*48+c, h, w]) * f[o,i,j]
__global__ __launch_bounds__(256) void idwt_haar_tdm_kernel(
    const float* __restrict__ x,
    const float* __restrict__ filt,
    float* __restrict__ out) {
  __shared__ float smem[4 * RPB * W_];   // 16 KB tile: [subband][row][col]

  // 3D grid: no integer division in the prologue.
  const int hblk = blockIdx.x;           // 0..15
  const int c    = blockIdx.y;           // 0..47
  const int b    = blockIdx.z;           // 0..7
  const int h0   = hblk * RPB;

  // 16 filter taps (4,2,2) -> registers BEFORE the barrier, so these loads
  // overlap the TDM wait instead of serializing after it.
  float fv[16];
#pragma unroll
  for (int k = 0; k < 16; ++k) fv[k] = filt[k];

  // ---- TDM: one 3D descriptor DMA (cols x rows x subbands) global -> LDS ----
  if ((threadIdx.x >> 5) == 0) {         // wave 0 only (uniform branch)
    const uint64_t gaddr =
        (uint64_t)(uintptr_t)(x + (((size_t)(b * 4 * C_ + c) * H_) + h0) * W_);
    const uint32_t laddr = (uint32_t)(uintptr_t)&smem[0];
    const uint64_t zstride = (uint64_t)C_ * H_ * W_;   // 786432 elems between subbands

    uint32x4 g0;
    g0[0] = 1u;                                            // count=1, user mode
    g0[1] = laddr;                                         // lds_addr (bytes)
    g0[2] = (uint32_t)gaddr;                               // global_addr[31:0]
    g0[3] = (uint32_t)((gaddr >> 32) & 0x1FFFFFFull)       // global_addr[56:32]
          | (2u << 30);                                    // type=2 ("image")

    int32x8 g1;
    g1[0] = (2 << 16);                   // data_size=4B; no multicast/pad/iterate
    g1[1] = (W_ << 16);                  // tensor_dim0[15:0]=128 (bits 63:48)
    g1[2] = (RPB << 16);                 // tensor_dim0 hi=0 | tensor_dim1[15:0]=8
    g1[3] = (W_ << 16);                  // tensor_dim1 hi=0 | tile_dim0=128
    g1[4] = RPB | (4 << 16);             // tile_dim1=8 | tile_dim2=4
    g1[5] = W_;                          // tensor_dim0_stride[31:0]=128
    g1[6] = (int)((zstride & 0xFFFFull) << 16);  // dim0_stride hi | dim1_stride[15:0]
    g1[7] = (int)(zstride >> 16);                // dim1_stride[47:16]

    int32x4 g2;
    g2[0] = 4;                           // tensor_dim2 = 4 subbands
    g2[1] = 0; g2[2] = 0; g2[3] = 0;     // no dim3 / iterate
    int32x4 g3 = {0, 0, 0, 0};

#if defined(__clang_major__) && (__clang_major__ >= 23)
    int32x8 g4 = {0, 0, 0, 0, 0, 0, 0, 0};
    __builtin_amdgcn_tensor_load_to_lds(g0, g1, g2, g3, g4, 0);
#else
    __builtin_amdgcn_tensor_load_to_lds(g0, g1, g2, g3, 0);
#endif
    __builtin_amdgcn_s_wait_tensorcnt(0);
  }
  __syncthreads();

  // 512 tasks (8 rows x 64 col-pairs), exactly 2 per thread -- unrolled.
#pragma unroll
  for (int k = 0; k < 2; ++k) {
    const int t  = threadIdx.x + k * 256;
    const int r  = t >> 6;
    const int wp = t & 63;
    const int wi = 2 * wp;

    v2f a0 = *(const v2f*)&smem[(0 * RPB + r) * W_ + wi];
    v2f a1 = *(const v2f*)&smem[(1 * RPB + r) * W_ + wi];
    v2f a2 = *(const v2f*)&smem[(2 * RPB + r) * W_ + wi];
    v2f a3 = *(const v2f*)&smem[(3 * RPB + r) * W_ + wi];
    const float v0 = a0[0] + a1[0] + a2[0] + a3[0];
    const float v1 = a0[1] + a1[1] + a2[1] + a3[1];

    const int orow = 2 * (h0 + r);
    const int ocol = 4 * wp;
#pragma unroll
    for (int o = 0; o < 4; ++o) {
#pragma unroll
      for (int i = 0; i < 2; ++i) {
        const float f0 = fv[o * 4 + i * 2 + 0];
        const float f1 = fv[o * 4 + i * 2 + 1];
        v4f vec = { v0 * f0, v0 * f1, v1 * f0, v1 * f1 };
        const size_t idx =
            (((size_t)b * (4 * C_) + (4 * c + o)) * (2 * H_) + (orow + i))
                * (size_t)(2 * W_) + ocol;
        __builtin_nontemporal_store(vec, (v4f*)&out[idx]);   // streamed 403 MB
      }
    }
  }
}

extern "C" void kernel_launch(void* const* d_in, const int* in_sizes, int n_in,
                              void* d_out, int out_size, void* d_ws, size_t ws_size,
                              hipStream_t stream) {
  const float* x    = (const float*)d_in[0];
  const float* filt = (const float*)d_in[1];
  float*       out  = (float*)d_out;
  (void)in_sizes; (void)n_in; (void)out_size; (void)d_ws; (void)ws_size;

  dim3 grid(H_ / RPB, C_, B_);           // (16, 48, 8) = 6144 blocks, 8 waves each
  idwt_haar_tdm_kernel<<<grid, 256, 0, stream>>>(x, filt, out);
}